// Net_16801912062046
// MI455X (gfx1250) — compile-verified
//
#include <hip/hip_runtime.h>
#include <math.h>

typedef __attribute__((ext_vector_type(2))) float v2f;
typedef __attribute__((ext_vector_type(8))) float v8f;

#define N_NODES   100000
#define N_EDGES   1600000
#define IN_FEAT   256
#define HIDDEN    128
#define N_CLASSES 16

// ---------------- degree / normalization ----------------

__global__ void init_deg_kernel(float* __restrict__ deg) {
    int i = blockIdx.x * blockDim.x + threadIdx.x;
    if (i < N_NODES) deg[i] = 1.0f;   // self-loop contributes 1
}

__global__ void deg_accum_kernel(const int* __restrict__ col, float* __restrict__ deg) {
    int e = blockIdx.x * blockDim.x + threadIdx.x;
    if (e < N_EDGES) unsafeAtomicAdd(&deg[col[e]], 1.0f);
}

__global__ void dinv_kernel(float* __restrict__ deg) {
    int i = blockIdx.x * blockDim.x + threadIdx.x;
    if (i < N_NODES) deg[i] = rsqrtf(deg[i]);   // deg >= 1 always
}

// ---------------- GEMM1: hp = x @ W1  (f32 WMMA 16x16x4) ----------------
// One wave computes rows [row0, row0+16) x all 128 output cols (8 tiles).

__global__ __launch_bounds__(256) void gemm1_kernel(const float* __restrict__ x,
                                                    const float* __restrict__ W1,
                                                    float* __restrict__ hp) {
    const int lane = threadIdx.x & 31;
    const int wave = (blockIdx.x * blockDim.x + threadIdx.x) >> 5;
    const int row0 = wave * 16;
    if (row0 >= N_NODES) return;            // wave-uniform: EXEC all-1s inside

    const int m  = lane & 15;               // M (for A) / N (for B) within tile
    const int kh = lane >> 4;               // selects K pair {2*kh, 2*kh+1}
    const float* arow = x + (size_t)(row0 + m) * IN_FEAT;

    v8f acc[8];
#pragma unroll
    for (int n = 0; n < 8; ++n) acc[n] = v8f{};

    for (int k = 0; k < IN_FEAT; k += 4) {
        v2f a;
        a.x = arow[k + 2 * kh + 0];
        a.y = arow[k + 2 * kh + 1];
        const float* b0 = W1 + (size_t)(k + 2 * kh) * HIDDEN + m;
#pragma unroll
        for (int n = 0; n < 8; ++n) {
            v2f b;
            b.x = b0[n * 16];
            b.y = b0[HIDDEN + n * 16];
            acc[n] = __builtin_amdgcn_wmma_f32_16x16x4_f32(
                false, a, false, b, (short)0, acc[n], false, false);
        }
    }

    const int mo = kh * 8;                  // lanes 16-31 hold rows M=8..15
#pragma unroll
    for (int n = 0; n < 8; ++n) {
#pragma unroll
        for (int j = 0; j < 8; ++j) {
            hp[(size_t)(row0 + mo + j) * HIDDEN + n * 16 + m] = acc[n][j];
        }
    }
}

// ---------------- GEMM2: h2p = h @ W2  (K=128, N=16) ----------------

__global__ __launch_bounds__(256) void gemm2_kernel(const float* __restrict__ h,
                                                    const float* __restrict__ W2,
                                                    float* __restrict__ h2p) {
    const int lane = threadIdx.x & 31;
    const int wave = (blockIdx.x * blockDim.x + threadIdx.x) >> 5;
    const int row0 = wave * 16;
    if (row0 >= N_NODES) return;

    const int m  = lane & 15;
    const int kh = lane >> 4;
    const float* arow = h + (size_t)(row0 + m) * HIDDEN;

    v8f acc = v8f{};
    for (int k = 0; k < HIDDEN; k += 4) {
        v2f a;
        a.x = arow[k + 2 * kh + 0];
        a.y = arow[k + 2 * kh + 1];
        v2f b;
        b.x = W2[(size_t)(k + 2 * kh + 0) * N_CLASSES + m];
        b.y = W2[(size_t)(k + 2 * kh + 1) * N_CLASSES + m];
        acc = __builtin_amdgcn_wmma_f32_16x16x4_f32(
            false, a, false, b, (short)0, acc, false, false);
    }

    const int mo = kh * 8;
#pragma unroll
    for (int j = 0; j < 8; ++j) {
        h2p[(size_t)(row0 + mo + j) * N_CLASSES + m] = acc[j];
    }
}

// ---------------- aggregation ----------------

// agg[i*F + f] = dinv[i]^2 * src[i*F + f]   (self-loop term; also initializes agg)
__global__ void agg_self_kernel(const float* __restrict__ src,
                                const float* __restrict__ dinv,
                                float* __restrict__ agg,
                                int shift, long total) {
    long i = (long)blockIdx.x * blockDim.x + threadIdx.x;
    if (i < total) {
        int node = (int)(i >> shift);
        float d = dinv[node];
        agg[i] = d * d * src[i];
    }
}

// layer-1 edges: one wave per edge, 4 feats per lane (128 feats)
__global__ __launch_bounds__(256) void agg1_edge_kernel(const int* __restrict__ row,
                                                        const int* __restrict__ col,
                                                        const float* __restrict__ dinv,
                                                        const float* __restrict__ hp,
                                                        float* __restrict__ agg) {
    const int lane = threadIdx.x & 31;
    const long e = ((long)blockIdx.x * blockDim.x + threadIdx.x) >> 5;
    if (e >= N_EDGES) return;
    const int r = row[e];
    const int c = col[e];
    const float w = dinv[r] * dinv[c];
    const float4 v = ((const float4*)(hp + (size_t)r * HIDDEN))[lane];
    float* dst = agg + (size_t)c * HIDDEN + lane * 4;
    unsafeAtomicAdd(dst + 0, w * v.x);
    unsafeAtomicAdd(dst + 1, w * v.y);
    unsafeAtomicAdd(dst + 2, w * v.z);
    unsafeAtomicAdd(dst + 3, w * v.w);
}

// layer-2 edges: one thread per (edge, feature), 16 feats
__global__ void agg2_edge_kernel(const int* __restrict__ row,
                                 const int* __restrict__ col,
                                 const float* __restrict__ dinv,
                                 const float* __restrict__ h2p,
                                 float* __restrict__ agg) {
    long t = (long)blockIdx.x * blockDim.x + threadIdx.x;
    if (t >= (long)N_EDGES * N_CLASSES) return;
    int e = (int)(t >> 4);
    int f = (int)(t & 15);
    int r = row[e];
    int c = col[e];
    float w = dinv[r] * dinv[c];
    unsafeAtomicAdd(&agg[(size_t)c * N_CLASSES + f], w * h2p[(size_t)r * N_CLASSES + f]);
}

// out[i*128+f] = relu(agg[i*128+f] + b1[f])
__global__ void bias_relu_kernel(const float* __restrict__ agg,
                                 const float* __restrict__ b1,
                                 float* __restrict__ out) {
    long i = (long)blockIdx.x * blockDim.x + threadIdx.x;
    if (i < (long)N_NODES * HIDDEN) {
        float v = agg[i] + b1[(int)(i & (HIDDEN - 1))];
        out[i] = v > 0.0f ? v : 0.0f;
    }
}

// log_softmax over 16 classes (with bias b2 added first)
__global__ void logsoftmax_kernel(const float* __restrict__ agg,
                                  const float* __restrict__ b2,
                                  float* __restrict__ out) {
    int i = blockIdx.x * blockDim.x + threadIdx.x;
    if (i >= N_NODES) return;
    float v[N_CLASSES];
    float mx = -INFINITY;
#pragma unroll
    for (int c = 0; c < N_CLASSES; ++c) {
        v[c] = agg[(size_t)i * N_CLASSES + c] + b2[c];
        mx = fmaxf(mx, v[c]);
    }
    float s = 0.0f;
#pragma unroll
    for (int c = 0; c < N_CLASSES; ++c) s += expf(v[c] - mx);
    float lse = mx + logf(s);
#pragma unroll
    for (int c = 0; c < N_CLASSES; ++c) {
        out[(size_t)i * N_CLASSES + c] = v[c] - lse;
    }
}

// ---------------- launch ----------------

static inline int cdiv_l(long a, int b) { return (int)((a + b - 1) / b); }

extern "C" void kernel_launch(void* const* d_in, const int* in_sizes, int n_in,
                              void* d_out, int out_size, void* d_ws, size_t ws_size,
                              hipStream_t stream) {
    const float* x   = (const float*)d_in[0];
    const int*   ei  = (const int*)d_in[1];
    const int*   row = ei;              // edge_index[0] = source
    const int*   col = ei + N_EDGES;    // edge_index[1] = destination
    const float* W1  = (const float*)d_in[2];
    const float* b1  = (const float*)d_in[3];
    const float* W2  = (const float*)d_in[4];
    const float* b2  = (const float*)d_in[5];
    float* out = (float*)d_out;

    float* ws   = (float*)d_ws;
    float* dinv = ws;                                        // N (deg, then rsqrt in place)
    float* hp   = ws + ((N_NODES + 255) & ~255);             // N*128 proj; later relu output
    float* agg1 = hp + (size_t)N_NODES * HIDDEN;             // N*128 aggregate
    float* h2p  = agg1;                                      // N*16 (agg1 dead by then)
    float* agg2 = agg1 + (size_t)N_NODES * N_CLASSES;        // N*16

    const int B = 256;

    // normalization
    init_deg_kernel<<<cdiv_l(N_NODES, B), B, 0, stream>>>(dinv);
    deg_accum_kernel<<<cdiv_l(N_EDGES, B), B, 0, stream>>>(col, dinv);
    dinv_kernel<<<cdiv_l(N_NODES, B), B, 0, stream>>>(dinv);

    // layer 1
    const long waves1 = (N_NODES + 15) / 16;                 // 6250 waves
    gemm1_kernel<<<cdiv_l(waves1 * 32, B), B, 0, stream>>>(x, W1, hp);
    agg_self_kernel<<<cdiv_l((long)N_NODES * HIDDEN, B), B, 0, stream>>>(
        hp, dinv, agg1, 7, (long)N_NODES * HIDDEN);
    agg1_edge_kernel<<<cdiv_l((long)N_EDGES * 32, B), B, 0, stream>>>(
        row, col, dinv, hp, agg1);
    bias_relu_kernel<<<cdiv_l((long)N_NODES * HIDDEN, B), B, 0, stream>>>(agg1, b1, hp);

    // layer 2
    gemm2_kernel<<<cdiv_l(waves1 * 32, B), B, 0, stream>>>(hp, W2, h2p);
    agg_self_kernel<<<cdiv_l((long)N_NODES * N_CLASSES, B), B, 0, stream>>>(
        h2p, dinv, agg2, 4, (long)N_NODES * N_CLASSES);
    agg2_edge_kernel<<<cdiv_l((long)N_EDGES * N_CLASSES, B), B, 0, stream>>>(
        row, col, dinv, h2p, agg2);

    // output
    logsoftmax_kernel<<<cdiv_l(N_NODES, B), B, 0, stream>>>(agg2, b2, out);
}